// BottleNeck_70600672411880
// MI455X (gfx1250) — compile-verified
//
#include <hip/hip_runtime.h>
#include <hip/hip_bf16.h>
#include <math.h>

// ---------------------------------------------------------------------------
// Problem constants (from the reference)
// ---------------------------------------------------------------------------
#define BATCH   8
#define C_IN    512
#define HH      32
#define WW      32
#define HW      1024          // 32*32
#define PLANES  128
#define GROUPS  8
#define DD      16            // PLANES / GROUPS
#define KS      7
#define PADW    3
#define PH      38            // H + 2*PAD
#define PHW     1444          // 38*38
#define C_OUT   512           // PLANES*4

typedef float        v2f __attribute__((ext_vector_type(2)));
typedef float        v8f __attribute__((ext_vector_type(8)));
typedef unsigned int v4u __attribute__((ext_vector_type(4)));
typedef int          v4i __attribute__((ext_vector_type(4)));
typedef int          v8i __attribute__((ext_vector_type(8)));

// ---------------------------------------------------------------------------
// WMMA fragment conventions (wave32, V_WMMA_F32_16X16X4_F32):
//   A (16x4):  lane L holds M = L%16, K = {2*(L/16), 2*(L/16)+1}
//   B (4x16):  lane L holds N = L%16, K = {2*(L/16), 2*(L/16)+1}
//   C/D:       vgpr j, lane L holds M = j + 8*(L/16), N = L%16
// ---------------------------------------------------------------------------

// Generic channel-major GEMM:  O[b, o, hw] = sum_c A[b, c, hw] * Wm[o, c]
// A is (BATCH, K, HW) fp32 channel-major; Wm is (Nch, K) row-major.
// The 16(pos) x K(ch) A tile is DMA'd into LDS by the Tensor Data Mover:
// a 2D tile, row = 16 contiguous floats (positions), K rows, row stride HW.
// TDM packs rows contiguously in LDS -> sA layout is [c][16].
template <int K>
__global__ __launch_bounds__(256) void gemm_wmma_kernel(
    const float* __restrict__ A, const float* __restrict__ Wm,
    float* __restrict__ O, int Nch) {
  __shared__ float sA[16 * K];   // only LDS object -> segment offset 0
  const int tid  = threadIdx.x;
  const int lane = tid & 31;
  const int wid  = tid >> 5;
  const int p0   = blockIdx.x * 16;
  const int b    = p0 >> 10;         // HW == 1024
  const int hw0  = p0 & 1023;

  const float* Ab = A + (size_t)b * K * HW + hw0;

  if (wid == 0) {
    // ---- Tensor DMA descriptor (D#) ----
    const unsigned long long ga = (unsigned long long)(const void*)Ab;
    v4u g0;
    g0[0] = 1u;                                   // count=1, user mode
    g0[1] = 0u;                                   // lds_addr = 0 (sA base)
    g0[2] = (unsigned)(ga & 0xFFFFFFFFu);         // global_addr[31:0]
    g0[3] = (unsigned)((ga >> 32) & 0x01FFFFFFu)  // global_addr[56:32]
            | (2u << 30);                         // type = 2 (image)
    v8i g1;
    g1[0] = (2 << 16);        // data_size = 4 bytes; no multicast/pad/iterate
    g1[1] = (16 << 16);       // abar_addr=0 | tensor_dim0 = 16
    g1[2] = (K << 16);        // tensor_dim0[31:16]=0 | tensor_dim1 = K
    g1[3] = (16 << 16);       // tensor_dim1[31:16]=0 | tile_dim0 = 16
    g1[4] = K;                // tile_dim1 = K | tile_dim2 = 0
    g1[5] = HW;               // tensor_dim0_stride[31:0] = 1024 elements
    g1[6] = 0;                // stride0[47:32]=0 | tensor_dim1_stride lo = 0
    g1[7] = 0;
    v4i z4 = {};
#if defined(__clang_major__) && (__clang_major__ >= 23)
    v8i z8 = {};
    __builtin_amdgcn_tensor_load_to_lds(g0, g1, z4, z4, z8, 0);
#else
    __builtin_amdgcn_tensor_load_to_lds(g0, g1, z4, z4, 0);
#endif
    __builtin_amdgcn_s_wait_tensorcnt(0);
  }
  __syncthreads();

  const int ntile = blockIdx.y * 8 + wid;
  const int o0    = ntile * 16;
  const int r     = lane & 15;
  const int hi    = lane >> 4;

  v8f acc = {};
  const float* Wrow = Wm + (size_t)(o0 + r) * K;
#pragma unroll 8
  for (int k = 0; k < K; k += 4) {
    const int ka = k + 2 * hi;
    v2f a, bm;
    a[0]  = sA[ka * 16 + r];          // A(M=r, K=ka)   [c][16] layout
    a[1]  = sA[(ka + 1) * 16 + r];    // A(M=r, K=ka+1)
    bm[0] = Wrow[ka];
    bm[1] = Wrow[ka + 1];
    acc = __builtin_amdgcn_wmma_f32_16x16x4_f32(false, a, false, bm,
                                                (short)0, acc, false, false);
  }

  float* Ob = O + (size_t)b * Nch * HW + (size_t)(o0 + r) * HW + hw0;
#pragma unroll
  for (int j = 0; j < 8; ++j) Ob[j + 8 * hi] = acc[j];
}

// q projection: per-group 16x16x16 GEMM over 16-position tiles.
// q[b, g*16+o, hw] = sum_i t1[b, g*16+i, hw] * Wq[g, o, i]
__global__ __launch_bounds__(256) void qproj_kernel(
    const float* __restrict__ t1, const float* __restrict__ Wq,
    float* __restrict__ qp) {
  __shared__ float sA[16 * 129];
  const int tid  = threadIdx.x;
  const int lane = tid & 31;
  const int g    = tid >> 5;       // wave id == group
  const int p0   = blockIdx.x * 16;
  const int b    = p0 >> 10;
  const int hw0  = p0 & 1023;

  const float* Ab = t1 + (size_t)b * PLANES * HW + hw0;
  for (int idx = tid; idx < 16 * PLANES; idx += 256) {
    int p = idx & 15, c = idx >> 4;
    sA[p * 129 + c] = Ab[(size_t)c * HW + p];
  }
  __syncthreads();

  const int r  = lane & 15;
  const int hi = lane >> 4;
  v8f acc = {};
  const float* Wg = Wq + g * (DD * DD) + r * DD;   // row o=r of Wq[g]
  const float* sArow = sA + r * 129 + g * DD;
#pragma unroll
  for (int k = 0; k < DD; k += 4) {
    const int ka = k + 2 * hi;
    v2f a, bm;
    a[0]  = sArow[ka];
    a[1]  = sArow[ka + 1];
    bm[0] = Wg[ka];
    bm[1] = Wg[ka + 1];
    acc = __builtin_amdgcn_wmma_f32_16x16x4_f32(false, a, false, bm,
                                                (short)0, acc, false, false);
  }
  float* Ob = qp + (size_t)b * PLANES * HW + (size_t)(g * DD + r) * HW + hw0;
#pragma unroll
  for (int j = 0; j < 8; ++j) Ob[j + 8 * hi] = acc[j];
}

// k/v projections over the zero-padded 38x38 domain, biases added.
// kp/vp layout: (BATCH, PLANES, PHW)
__global__ __launch_bounds__(256) void kvproj_kernel(
    const float* __restrict__ t1,
    const float* __restrict__ Wk, const float* __restrict__ bk,
    const float* __restrict__ Wv, const float* __restrict__ bv,
    float* __restrict__ kp, float* __restrict__ vp) {
  __shared__ float sA[16 * 129];
  const int tid  = threadIdx.x;
  const int lane = tid & 31;
  const int g    = tid >> 5;
  const int pp0  = blockIdx.x * 16;   // 722 tiles * 16 == BATCH*PHW exactly

  for (int idx = tid; idx < 16 * PLANES; idx += 256) {
    int p = idx & 15, c = idx >> 4;
    int pp  = pp0 + p;
    int b   = pp / PHW;
    int rem = pp - b * PHW;
    int y   = rem / PH - PADW;
    int x   = rem - (y + PADW) * PH - PADW;
    float v = 0.f;
    if ((unsigned)y < (unsigned)HH && (unsigned)x < (unsigned)WW)
      v = t1[(size_t)b * PLANES * HW + (size_t)c * HW + y * WW + x];
    sA[p * 129 + c] = v;
  }
  __syncthreads();

  const int r  = lane & 15;
  const int hi = lane >> 4;
  v8f ak = {}, av = {};
  const float* WgK = Wk + g * (DD * DD) + r * DD;
  const float* WgV = Wv + g * (DD * DD) + r * DD;
  const float* sArow = sA + r * 129 + g * DD;
#pragma unroll
  for (int k = 0; k < DD; k += 4) {
    const int ka = k + 2 * hi;
    v2f a, bmk, bmv;
    a[0]   = sArow[ka];
    a[1]   = sArow[ka + 1];
    bmk[0] = WgK[ka];
    bmk[1] = WgK[ka + 1];
    bmv[0] = WgV[ka];
    bmv[1] = WgV[ka + 1];
    ak = __builtin_amdgcn_wmma_f32_16x16x4_f32(false, a, false, bmk,
                                               (short)0, ak, false, false);
    av = __builtin_amdgcn_wmma_f32_16x16x4_f32(false, a, false, bmv,
                                               (short)0, av, false, false);
  }
  const float biask = bk[g * DD + r];
  const float biasv = bv[g * DD + r];
#pragma unroll
  for (int j = 0; j < 8; ++j) {
    int pp  = pp0 + j + 8 * hi;
    int b   = pp / PHW;
    int rem = pp - b * PHW;
    size_t o = (size_t)b * PLANES * PHW + (size_t)(g * DD + r) * PHW + rem;
    kp[o] = ak[j] + biask;
    vp[o] = av[j] + biasv;
  }
}

// Online-softmax attention: one thread per (b, channel, h, w); 49 taps.
__global__ __launch_bounds__(256) void attention_kernel(
    const float* __restrict__ qp, const float* __restrict__ kp,
    const float* __restrict__ vp, const float* __restrict__ relx,
    const float* __restrict__ rely, float* __restrict__ att) {
  const int gid = blockIdx.x * 256 + threadIdx.x;   // BATCH*PLANES*HW threads
  const int hw = gid & 1023;
  const int ch = (gid >> 10) & (PLANES - 1);
  const int b  = gid >> 17;
  const int d  = ch & (DD - 1);
  const int h  = hw >> 5;
  const int w  = hw & 31;

  const float  q    = qp[gid];
  const size_t base = (size_t)b * PLANES * PHW + (size_t)ch * PHW;

  float m = -INFINITY, su = 0.f, acc = 0.f;
#pragma unroll
  for (int ky = 0; ky < KS; ++ky) {
    const size_t row = base + (size_t)(h + ky) * PH + w;
    const float rx = (d < 8) ? relx[d * KS + ky] : 0.f;
#pragma unroll
    for (int kx = 0; kx < KS; ++kx) {
      const float rel = (d < 8) ? rx : rely[(d - 8) * KS + kx];
      const float kk  = kp[row + kx];
      const float vv  = vp[row + kx];
      const float l   = q * (kk + rel);
      const float nm  = fmaxf(m, l);
      const float sc  = __expf(m - nm);
      const float e   = __expf(l - nm);
      su  = su * sc + e;
      acc = acc * sc + e * vv;
      m   = nm;
    }
  }
  att[gid] = acc / su;
}

// Per-channel mean / inv-std over (B, HW) for a (B, C, HW) tensor.
__global__ __launch_bounds__(256) void bn_stats_kernel(
    const float* __restrict__ src, int C,
    float* __restrict__ mean, float* __restrict__ istd) {
  __shared__ float ssum[256], ssq[256];
  const int c = blockIdx.x, tid = threadIdx.x;
  float s = 0.f, sq = 0.f;
  for (int i = tid; i < BATCH * HW; i += 256) {
    const int b = i >> 10, hw = i & 1023;
    const float v = src[((size_t)b * C + c) * HW + hw];
    s += v; sq += v * v;
  }
  ssum[tid] = s; ssq[tid] = sq;
  __syncthreads();
  for (int st = 128; st > 0; st >>= 1) {
    if (tid < st) { ssum[tid] += ssum[tid + st]; ssq[tid] += ssq[tid + st]; }
    __syncthreads();
  }
  if (tid == 0) {
    const float inv = 1.f / (float)(BATCH * HW);
    const float mu  = ssum[0] * inv;
    const float var = ssq[0] * inv - mu * mu;
    mean[c] = mu;
    istd[c] = rsqrtf(var + 1e-5f);
  }
}

// In-place BN apply + ReLU.
__global__ __launch_bounds__(256) void bn_apply_relu_kernel(
    float* __restrict__ buf, const float* __restrict__ mean,
    const float* __restrict__ istd, const float* __restrict__ gamma,
    const float* __restrict__ beta, int C, int total) {
  const int gid = blockIdx.x * 256 + threadIdx.x;
  if (gid >= total) return;
  const int ch = (gid >> 10) % C;
  const float v = buf[gid];
  const float r = (v - mean[ch]) * istd[ch] * gamma[ch] + beta[ch];
  buf[gid] = fmaxf(r, 0.f);
}

// Final: out = relu(bn3(out) + identity), in place on d_out.
__global__ __launch_bounds__(256) void bn_residual_relu_kernel(
    float* __restrict__ out, const float* __restrict__ x,
    const float* __restrict__ mean, const float* __restrict__ istd,
    const float* __restrict__ gamma, const float* __restrict__ beta) {
  const int gid = blockIdx.x * 256 + threadIdx.x;
  const int ch = (gid >> 10) & (C_OUT - 1);
  const float v = out[gid];
  const float r = (v - mean[ch]) * istd[ch] * gamma[ch] + beta[ch] + x[gid];
  out[gid] = fmaxf(r, 0.f);
}

// ---------------------------------------------------------------------------
extern "C" void kernel_launch(void* const* d_in, const int* in_sizes, int n_in,
                              void* d_out, int out_size, void* d_ws,
                              size_t ws_size, hipStream_t stream) {
  const float* x     = (const float*)d_in[0];
  const float* W1    = (const float*)d_in[1];
  const float* g1    = (const float*)d_in[2];
  const float* b1    = (const float*)d_in[3];
  const float* Wq    = (const float*)d_in[4];
  const float* Wk    = (const float*)d_in[5];
  const float* bk    = (const float*)d_in[6];
  const float* Wv    = (const float*)d_in[7];
  const float* bv    = (const float*)d_in[8];
  const float* rel_x = (const float*)d_in[9];
  const float* rel_y = (const float*)d_in[10];
  const float* g2    = (const float*)d_in[11];
  const float* b2    = (const float*)d_in[12];
  const float* W3    = (const float*)d_in[13];
  const float* g3    = (const float*)d_in[14];
  const float* b3    = (const float*)d_in[15];
  float* out = (float*)d_out;

  // Workspace layout (floats)
  float* ws  = (float*)d_ws;
  float* t1  = ws;                               // 8*128*1024   = 1,048,576
  float* qp  = t1 + (size_t)BATCH * PLANES * HW; // 1,048,576
  float* kp  = qp + (size_t)BATCH * PLANES * HW; // 8*128*1444   = 1,478,656
  float* vp  = kp + (size_t)BATCH * PLANES * PHW;
  float* att = vp + (size_t)BATCH * PLANES * PHW; // 1,048,576
  float* st  = att + (size_t)BATCH * PLANES * HW;
  float *mean1 = st,        *istd1 = st + 128;
  float *mean2 = st + 256,  *istd2 = st + 384;
  float *mean3 = st + 512,  *istd3 = st + 1024;

  const int Mtiles = (BATCH * HW) / 16;   // 512

  // 1) out1 = x @ W1^T  (TDM-staged A tile + WMMA f32, K=512, N=128)
  gemm_wmma_kernel<C_IN><<<dim3(Mtiles, 1), 256, 0, stream>>>(x, W1, t1, PLANES);
  // 2) BN1 + ReLU (in place)
  bn_stats_kernel<<<PLANES, 256, 0, stream>>>(t1, PLANES, mean1, istd1);
  bn_apply_relu_kernel<<<(BATCH * PLANES * HW) / 256, 256, 0, stream>>>(
      t1, mean1, istd1, g1, b1, PLANES, BATCH * PLANES * HW);
  // 3) q/k/v group projections (WMMA f32 16x16x16 per group)
  qproj_kernel<<<Mtiles, 256, 0, stream>>>(t1, Wq, qp);
  kvproj_kernel<<<(BATCH * PHW) / 16, 256, 0, stream>>>(t1, Wk, bk, Wv, bv, kp, vp);
  // 4) neighborhood softmax attention
  attention_kernel<<<(BATCH * PLANES * HW) / 256, 256, 0, stream>>>(
      qp, kp, vp, rel_x, rel_y, att);
  // 5) BN2 + ReLU (in place)
  bn_stats_kernel<<<PLANES, 256, 0, stream>>>(att, PLANES, mean2, istd2);
  bn_apply_relu_kernel<<<(BATCH * PLANES * HW) / 256, 256, 0, stream>>>(
      att, mean2, istd2, g2, b2, PLANES, BATCH * PLANES * HW);
  // 6) out3 = t2 @ W3^T  (TDM-staged A tile + WMMA f32, K=128, N=512) -> d_out
  gemm_wmma_kernel<PLANES><<<dim3(Mtiles, 4), 256, 0, stream>>>(att, W3, out, C_OUT);
  // 7) BN3 + residual + ReLU (in place on d_out)
  bn_stats_kernel<<<C_OUT, 256, 0, stream>>>(out, C_OUT, mean3, istd3);
  bn_residual_relu_kernel<<<(BATCH * C_OUT * HW) / 256, 256, 0, stream>>>(
      out, x, mean3, istd3, g3, b3);
}